// LocalGraphProjection_81123342286852
// MI455X (gfx1250) — compile-verified
//
#include <hip/hip_runtime.h>
#include <math.h>

#define NPTS   30000
#define NVIEWS 3
#define CTOT   1008
#define H0C    448
#define W0C    256

typedef float v2f __attribute__((ext_vector_type(2)));
typedef float v4f __attribute__((ext_vector_type(4)));
typedef float v8f __attribute__((ext_vector_type(8)));

__device__ __forceinline__ v4f vmax4(v4f a, v4f b) {
    v4f r;
    r.x = fmaxf(a.x, b.x); r.y = fmaxf(a.y, b.y);
    r.z = fmaxf(a.z, b.z); r.w = fmaxf(a.w, b.w);
    return r;
}
__device__ __forceinline__ v4f vsqrt4(v4f a) {
    v4f r;
    r.x = sqrtf(a.x); r.y = sqrtf(a.y);
    r.z = sqrtf(a.z); r.w = sqrtf(a.w);
    return r;
}

// ---------------------------------------------------------------------------
// Setup: fold (inputs - t0) @ R0 @ Rv^T + tv into one homogeneous 4x16 B
// operand (columns 3v+j hold the affine for view v, coord j), plus folded
// intrinsics (fy/ratio_h etc). Tiny; one thread.
// ---------------------------------------------------------------------------
__global__ void setup_kernel(const float* __restrict__ ext,
                             const float* __restrict__ intr,
                             const int*   __restrict__ sv,
                             float* __restrict__ wsB,     // 4x16 row-major
                             float* __restrict__ wsProj)  // 3 views x {fyr,pyr,fxr,pxr}
{
    if (threadIdx.x != 0 || blockIdx.x != 0) return;
    float R0[3][3], t0[3];
    for (int m = 0; m < 3; ++m) {
        for (int k = 0; k < 3; ++k) R0[m][k] = ext[m * 4 + k];
        t0[m] = ext[m * 4 + 3];
    }
    for (int i = 0; i < 64; ++i) wsB[i] = 0.0f;
    const float ratio_h = 1280.0f / 448.0f;
    const float ratio_w = 720.0f  / 256.0f;
    for (int v = 0; v < NVIEWS; ++v) {
        int cam = sv[v];
        float Rv[3][3], tv[3];
        for (int j = 0; j < 3; ++j) {
            for (int k = 0; k < 3; ++k) Rv[j][k] = ext[cam * 16 + j * 4 + k];
            tv[j] = ext[cam * 16 + j * 4 + 3];
        }
        for (int j = 0; j < 3; ++j) {
            float cconst = tv[j];
            for (int m = 0; m < 3; ++m) {
                float a = 0.0f;
                for (int k = 0; k < 3; ++k) a += R0[m][k] * Rv[j][k];
                wsB[m * 16 + (3 * v + j)] = a;   // A_v[m][j]
                cconst -= t0[m] * a;
            }
            wsB[3 * 16 + (3 * v + j)] = cconst;  // homogeneous row
        }
        float fx = intr[cam * 4 + 0], fy = intr[cam * 4 + 1];
        float px = intr[cam * 4 + 2], py = intr[cam * 4 + 3];
        wsProj[v * 4 + 0] = fy / ratio_h;
        wsProj[v * 4 + 1] = py / ratio_h;
        wsProj[v * 4 + 2] = fx / ratio_w;
        wsProj[v * 4 + 3] = px / ratio_w;
    }
}

// ---------------------------------------------------------------------------
// Transpose one pyramid level (V,C,H,W) -> (V,H,W,C) via LDS tiles so the
// gather kernel gets coalesced channel reads. One (v,x) plane per blockIdx.z.
// ---------------------------------------------------------------------------
__global__ void transpose_kernel(const float* __restrict__ in,
                                 float* __restrict__ out,
                                 int C, int H, int W)
{
    __shared__ float tile[32][33];
    const int y0 = blockIdx.x * 32;
    const int c0 = blockIdx.y * 32;
    const int vx = blockIdx.z;
    const int v  = vx / H;
    const int x  = vx % H;
    const size_t HW = (size_t)H * W;

    for (int i = threadIdx.y; i < 32; i += 8) {
        int c = c0 + i, y = y0 + threadIdx.x;
        if (c < C && y < W)
            tile[i][threadIdx.x] = in[((size_t)v * C + c) * HW + (size_t)x * W + y];
    }
    __syncthreads();
    for (int i = threadIdx.y; i < 32; i += 8) {
        int y = y0 + i, c = c0 + threadIdx.x;
        if (c < C && y < W)
            out[((size_t)v * H + x) * W * (size_t)C + (size_t)y * C + c] = tile[threadIdx.x][i];
    }
}

// ---------------------------------------------------------------------------
// Projection: one wave32 per 16 points. A = 16x4 [x,y,z,1], B = 4x16 folded
// affines (3 views in 9 columns). D = 16x16 via V_WMMA_F32_16X16X4_F32, then
// regroup lane-striped X/Y/Z through LDS and emit clipped (h,w) per (v,n).
// ---------------------------------------------------------------------------
__global__ void project_kernel(const float* __restrict__ inputs,
                               const float* __restrict__ wsB,
                               const float* __restrict__ wsProj,
                               float* __restrict__ hw)  // [(v*NPTS+n)*2 + {0,1}]
{
    const int lane = threadIdx.x;            // 0..31
    const int base = blockIdx.x * 16;        // NPTS divisible by 16
    __shared__ float tile[16][17];
    __shared__ float proj[12];
    if (lane < 12) proj[lane] = wsProj[lane];

    // A layout (32-bit 16x4): lanes 0-15 hold K=0,1 ; lanes 16-31 hold K=2,3
    // B layout (4x16): symmetric striping — lanes 0-15 rows K=0,1 ; 16-31 rows K=2,3
    v2f a, b;
    if (lane < 16) {
        a.x = inputs[(base + lane) * 3 + 0];
        a.y = inputs[(base + lane) * 3 + 1];
        b.x = wsB[0 * 16 + lane];
        b.y = wsB[1 * 16 + lane];
    } else {
        a.x = inputs[(base + lane - 16) * 3 + 2];
        a.y = 1.0f;
        b.x = wsB[2 * 16 + (lane - 16)];
        b.y = wsB[3 * 16 + (lane - 16)];
    }
    v8f c = {};
    v8f d = __builtin_amdgcn_wmma_f32_16x16x4_f32(
        /*neg_a=*/false, a, /*neg_b=*/false, b,
        /*c_mod=*/(short)0, c, /*reuse_a=*/false, /*reuse_b=*/false);

    // D layout: VGPR r -> lanes 0-15: (M=r, N=lane); lanes 16-31: (M=8+r, N=lane-16)
#pragma unroll
    for (int r = 0; r < 8; ++r) {
        if (lane < 16) tile[r][lane] = d[r];
        else           tile[8 + r][lane - 16] = d[r];
    }
    __syncthreads();

    for (int task = lane; task < 48; task += 32) {
        int pt = task & 15;
        int v  = task >> 4;
        float X = tile[pt][3 * v + 0];
        float Y = tile[pt][3 * v + 1];
        float Z = tile[pt][3 * v + 2] + 1e-8f;
        float hh = proj[v * 4 + 0] * (Y / Z)    + proj[v * 4 + 1];
        float ww = proj[v * 4 + 2] * (X / (-Z)) + proj[v * 4 + 3];
        const float hup = 447.0f, wup = 255.0f;
        hh = isnan(hh) ? hup : hh;  hh = fminf(fmaxf(hh, 0.0f), hup);
        ww = isnan(ww) ? wup : ww;  ww = fminf(fmaxf(ww, 0.0f), wup);
        int n = base + pt;
        hw[((size_t)v * NPTS + n) * 2 + 0] = hh;
        hw[((size_t)v * NPTS + n) * 2 + 1] = ww;
    }
}

// ---------------------------------------------------------------------------
// Shared tap-parameter setup, common to both gather variants.
// ---------------------------------------------------------------------------
struct TapParams { int x1, x2, y1, y2; float w11, w21, w12, w22; };

__device__ __forceinline__ void compute_taps(const float* __restrict__ hw, int n,
                                             TapParams (*tp)[6], float* hv, float* wv)
{
    const int tid = threadIdx.x;
    if (tid < NVIEWS * 2) {
        int v = tid >> 1;
        float val = hw[((size_t)v * NPTS + n) * 2 + (tid & 1)];
        if ((tid & 1) == 0) hv[v] = val; else wv[v] = val;
    }
    __syncthreads();
    if (tid < NVIEWS * 6) {
        int v = tid / 6, l = tid % 6;
        int H = H0C >> l, W = W0C >> l;
        float inv = 1.0f / (float)(1 << l);
        float x = hv[v] * inv, y = wv[v] * inv;
        int x1 = (int)fminf(fmaxf(floorf(x), 0.0f), (float)(H - 1));
        int x2 = (int)fminf(fmaxf(ceilf(x),  0.0f), (float)(H - 1));
        int y1 = (int)fminf(fmaxf(floorf(y), 0.0f), (float)(W - 1));
        int y2 = (int)fminf(fmaxf(ceilf(y),  0.0f), (float)(W - 1));
        float x1f = (float)x1, x2f = (float)x2, y1f = (float)y1, y2f = (float)y2;
        TapParams p;
        p.x1 = x1; p.x2 = x2; p.y1 = y1; p.y2 = y2;
        p.w11 = (x2f - x) * (y2f - y);
        p.w21 = (x - x1f) * (y2f - y);
        p.w12 = (x2f - x) * (y - y1f);
        p.w22 = (x - x1f) * (y - y1f);
        tp[v][l] = p;
    }
    __syncthreads();
}

// ---------------------------------------------------------------------------
// Vectorized gather (transposed (V,H,W,C) layout): each thread owns a float4
// channel group -> 12x global_load_b128 taps, 3x global_store_b128 results.
// 252 vec4 groups per point = one pass of a 256-thread block.
// ---------------------------------------------------------------------------
__global__ void gather_kernel_vec(const float* __restrict__ hw,
                                  const float* __restrict__ f0, const float* __restrict__ f1,
                                  const float* __restrict__ f2, const float* __restrict__ f3,
                                  const float* __restrict__ f4, const float* __restrict__ f5,
                                  float* __restrict__ out)
{
    const int n   = blockIdx.x;
    const int tid = threadIdx.x;
    __shared__ TapParams tp[NVIEWS][6];
    __shared__ float hv[NVIEWS], wv[NVIEWS];
    compute_taps(hw, n, tp, hv, wv);

    const int c = tid * 4;                 // vec4 channel group
    if (c >= CTOT) return;
    int l, cl;
    const float* f;
    if      (c < 16)  { l = 0; cl = c;       f = f0; }
    else if (c < 48)  { l = 1; cl = c - 16;  f = f1; }
    else if (c < 112) { l = 2; cl = c - 48;  f = f2; }
    else if (c < 240) { l = 3; cl = c - 112; f = f3; }
    else if (c < 496) { l = 4; cl = c - 240; f = f4; }
    else              { l = 5; cl = c - 496; f = f5; }
    const int H = H0C >> l, W = W0C >> l, C = 16 << l;

    v4f vals[NVIEWS];
#pragma unroll
    for (int v = 0; v < NVIEWS; ++v) {
        TapParams p = tp[v][l];
        size_t vb = (size_t)v * H * W * C + cl;
        v4f t11 = *(const v4f*)(f + vb + ((size_t)p.x1 * W + p.y1) * C);
        v4f t21 = *(const v4f*)(f + vb + ((size_t)p.x2 * W + p.y1) * C);
        v4f t12 = *(const v4f*)(f + vb + ((size_t)p.x1 * W + p.y2) * C);
        v4f t22 = *(const v4f*)(f + vb + ((size_t)p.x2 * W + p.y2) * C);
        vals[v] = p.w11 * t11 + p.w21 * t21 + p.w12 * t12 + p.w22 * t22;
    }
    v4f vmax = vmax4(vals[0], vmax4(vals[1], vals[2]));
    v4f mean = (vals[0] + vals[1] + vals[2]) * (1.0f / 3.0f);
    v4f d0 = vals[0] - mean, d1 = vals[1] - mean, d2 = vals[2] - mean;
    v4f sd = vsqrt4((d0 * d0 + d1 * d1 + d2 * d2) * (1.0f / 3.0f));

    size_t ob = (size_t)n * (3 * CTOT);
    *(v4f*)(out + ob + c)            = vmax;
    *(v4f*)(out + ob + CTOT + c)     = mean;
    *(v4f*)(out + ob + 2 * CTOT + c) = sd;
}

// ---------------------------------------------------------------------------
// Scalar gather fallback for the original (V,C,H,W) layout (ws too small).
// ---------------------------------------------------------------------------
__global__ void gather_kernel_scalar(const float* __restrict__ hw,
                                     const float* __restrict__ f0, const float* __restrict__ f1,
                                     const float* __restrict__ f2, const float* __restrict__ f3,
                                     const float* __restrict__ f4, const float* __restrict__ f5,
                                     float* __restrict__ out)
{
    const int n   = blockIdx.x;
    const int tid = threadIdx.x;
    __shared__ TapParams tp[NVIEWS][6];
    __shared__ float hv[NVIEWS], wv[NVIEWS];
    compute_taps(hw, n, tp, hv, wv);

    for (int c = tid; c < CTOT; c += blockDim.x) {
        int l, cl;
        const float* f;
        if      (c < 16)  { l = 0; cl = c;       f = f0; }
        else if (c < 48)  { l = 1; cl = c - 16;  f = f1; }
        else if (c < 112) { l = 2; cl = c - 48;  f = f2; }
        else if (c < 240) { l = 3; cl = c - 112; f = f3; }
        else if (c < 496) { l = 4; cl = c - 240; f = f4; }
        else              { l = 5; cl = c - 496; f = f5; }
        const int H = H0C >> l, W = W0C >> l, C = 16 << l;

        float vals[NVIEWS];
#pragma unroll
        for (int v = 0; v < NVIEWS; ++v) {
            TapParams p = tp[v][l];
            size_t vb = ((size_t)v * C + cl) * (size_t)(H * W);
            float f11 = f[vb + (size_t)p.x1 * W + p.y1];
            float f21 = f[vb + (size_t)p.x2 * W + p.y1];
            float f12 = f[vb + (size_t)p.x1 * W + p.y2];
            float f22 = f[vb + (size_t)p.x2 * W + p.y2];
            vals[v] = p.w11 * f11 + p.w21 * f21 + p.w12 * f12 + p.w22 * f22;
        }
        float vmax = fmaxf(vals[0], fmaxf(vals[1], vals[2]));
        float mean = (vals[0] + vals[1] + vals[2]) * (1.0f / 3.0f);
        float d0 = vals[0] - mean, d1 = vals[1] - mean, d2 = vals[2] - mean;
        float sd = sqrtf((d0 * d0 + d1 * d1 + d2 * d2) * (1.0f / 3.0f));
        size_t ob = (size_t)n * (3 * CTOT);
        out[ob + c]            = vmax;
        out[ob + CTOT + c]     = mean;
        out[ob + 2 * CTOT + c] = sd;
    }
}

// ---------------------------------------------------------------------------
extern "C" void kernel_launch(void* const* d_in, const int* in_sizes, int n_in,
                              void* d_out, int out_size, void* d_ws, size_t ws_size,
                              hipStream_t stream)
{
    (void)in_sizes; (void)n_in; (void)out_size;
    const float* inputs = (const float*)d_in[0];
    const float* feats[6] = {
        (const float*)d_in[1], (const float*)d_in[2], (const float*)d_in[3],
        (const float*)d_in[4], (const float*)d_in[5], (const float*)d_in[6]};
    const float* ext  = (const float*)d_in[7];
    const float* intr = (const float*)d_in[8];
    const int*   sv   = (const int*)d_in[9];
    float* out = (float*)d_out;
    float* ws  = (float*)d_ws;

    // Workspace layout (floats): [transposed feats (6 levels)] [hw] [B] [proj]
    size_t foff[6], tot = 0;
    for (int l = 0; l < 6; ++l) {
        int H = H0C >> l, W = W0C >> l, C = 16 << l;
        foff[l] = tot;
        tot += (size_t)NVIEWS * C * H * W;
    }
    const size_t hw_elems = (size_t)NVIEWS * NPTS * 2;
    const size_t need_full = (tot + hw_elems + 64 + 12) * sizeof(float);
    const bool do_t = ws_size >= need_full;

    float* hwbuf  = ws + (do_t ? tot : 0);
    float* wsB    = hwbuf + hw_elems;
    float* wsProj = wsB + 64;

    setup_kernel<<<1, 32, 0, stream>>>(ext, intr, sv, wsB, wsProj);

    if (do_t) {
        for (int l = 0; l < 6; ++l) {
            int H = H0C >> l, W = W0C >> l, C = 16 << l;
            dim3 grid((W + 31) / 32, (C + 31) / 32, NVIEWS * H);
            transpose_kernel<<<grid, dim3(32, 8), 0, stream>>>(feats[l], ws + foff[l], C, H, W);
        }
    }

    project_kernel<<<NPTS / 16, 32, 0, stream>>>(inputs, wsB, wsProj, hwbuf);

    if (do_t) {
        gather_kernel_vec<<<NPTS, 256, 0, stream>>>(
            hwbuf,
            ws + foff[0], ws + foff[1], ws + foff[2],
            ws + foff[3], ws + foff[4], ws + foff[5],
            out);
    } else {
        gather_kernel_scalar<<<NPTS, 256, 0, stream>>>(
            hwbuf,
            feats[0], feats[1], feats[2], feats[3], feats[4], feats[5],
            out);
    }
}